// TNModel_47940424958388
// MI455X (gfx1250) — compile-verified
//
#include <hip/hip_runtime.h>

typedef __attribute__((ext_vector_type(2))) float v2f;
typedef __attribute__((ext_vector_type(8))) float v8f;

#define BATCH           65536
#define WAVES_PER_BLOCK 4
#define SAMP_PER_WAVE   16
#define THREADS         (WAVES_PER_BLOCK * 32)

// ---- LDS weight region layout (floats; complex stored as re,im pairs) ----
#define OFF_TOP   0        // top_w (4,4,2)       -> 64 floats
#define OFF_M1    64       // mids[0] (2,4,4,4)   -> 256
#define OFF_M2    320      // mids[1] (4,4,4,4)   -> 512
#define OFF_M3    832      // mids[2] (8,4,4,4)   -> 1024
#define OFF_M4    1856     // mids[3] (16,4,4,4)  -> 2048
#define OFF_M5    3904     // mids[4] (32,4,4,4)  -> 4096
#define OFF_M6    8000     // mids[5] (64,4,4,4)  -> 8192
#define OFF_LEAF  16192    // leaf_w (128,2,2,4)  -> 4096
#define W_TOTAL   20288

// ---- per-wave value buffers: [node][m:16][c:4][re,im] ----
#define NODE_STRIDE 128    // 16*4*2 floats per node
#define VW_SIZE     5120   // bufA(16 nodes)=2048 | bufB(8 nodes)=1024 | v16(16 nodes)=2048
#define BUFA        0
#define BUFB        2048
#define V16         3072

// One mid-layer step for one wave's 16-sample tile.
// v_out[m, j, c] = sum_{a,b} vL[m,a] * vR[m,b] * w[jBase+j, a, b, c]   (complex)
// done as a real GEMM [16 x 32] @ [32 x 8] via 8 accumulating f32 16x16x4 WMMAs.
__device__ __forceinline__ void apply_layer(const float* sw,   // LDS weights, 128 floats/node
                                            const float* src,  // LDS: 2*nOut nodes
                                            float*       dst,  // LDS: nOut nodes
                                            int nOut, int jBase, int lane)
{
    const int  m   = lane & 15;        // WMMA M row (sample) this lane feeds
    const int  h   = lane >> 4;        // selects K pair (A) / K rows (B)
    const bool act = (m < 8);          // columns 8..15 of B are zero padding
    const int  c   = (m >> 1) & 3;     // output bond index carried by this column
    const int  oc  = m & 1;            // 0 = real component column, 1 = imag

    for (int j = 0; j < nOut; ++j) {
        const float* wj = sw  + (jBase + j) * 128;          // w[j][a][b][c] complex
        const float* vL = src + (2 * j)     * NODE_STRIDE;
        const float* vR = src + (2 * j + 1) * NODE_STRIDE;
        v8f acc = {0.f, 0.f, 0.f, 0.f, 0.f, 0.f, 0.f, 0.f};
#pragma unroll
        for (int kk = 0; kk < 8; ++kk) {
            const int t  = 2 * kk + h;     // complex contraction index 0..15 (a*4+b)
            const int a  = t >> 2;
            const int b2 = t & 3;
            // A operand: lane holds re/im of u_t = vL[a]*vR[b] for sample m
            const float lre = vL[m * 8 + a * 2 + 0];
            const float lim = vL[m * 8 + a * 2 + 1];
            const float rre = vR[m * 8 + b2 * 2 + 0];
            const float rim = vR[m * 8 + b2 * 2 + 1];
            v2f A;
            A.x = lre * rre - lim * rim;   // re(u_t)  (K = 4kk + 2h)
            A.y = lre * rim + lim * rre;   // im(u_t)  (K = 4kk + 2h + 1)
            // B operand: column n=m, rows K=4kk+2h (re-row) and +1 (im-row)
            const float wre = wj[t * 8 + c * 2 + 0];
            const float wim = wj[t * 8 + c * 2 + 1];
            v2f B;
            B.x = act ? (oc ? wim :  wre) : 0.0f;   // re-row:  [re(w) | im(w)]
            B.y = act ? (oc ? wre : -wim) : 0.0f;   // im-row:  [-im(w) | re(w)]
            acc = __builtin_amdgcn_wmma_f32_16x16x4_f32(
                false, A, false, B, (short)0, acc, false, false);
        }
        if (act) {   // D: acc[r] = D[M=r+8h, N=m] -> v_out[sample r+8h][c], component oc
            float* dj = dst + j * NODE_STRIDE;
#pragma unroll
            for (int r = 0; r < 8; ++r)
                dj[(r + 8 * h) * 8 + c * 2 + oc] = acc[r];
        }
    }
}

// Leaf contraction: v[b, n, d] = sum_{p,q} x[b,2n,p]*x[b,2n+1,q]*leaf_w[n,p,q,d]
__device__ __forceinline__ void leaf_layer(const float* __restrict__ x,
                                           const float* swleaf, float* bufA,
                                           int bBase, int s, int lane)
{
    const int    m = lane & 15;
    const int    h = lane >> 4;
    const size_t b = (size_t)(bBase + m);
#pragma unroll
    for (int i = 0; i < 8; ++i) {
        const int nl = h * 8 + i;        // local leaf node 0..15 within subtree
        const int n  = s * 16 + nl;      // global leaf node 0..127
        // x[b, 2n, 0..1], x[b, 2n+1, 0..1] are 4 contiguous floats; the 8
        // i-iterations of one lane cover exactly one 128B cache line.
        const float4 xv = *(const float4*)(x + b * 512 + (size_t)n * 4);
        const float r00 = xv.x * xv.z;
        const float r01 = xv.x * xv.w;
        const float r10 = xv.y * xv.z;
        const float r11 = xv.y * xv.w;
        const float* wn = swleaf + n * 32;            // (p,q,d) complex
        float*       dn = bufA + nl * NODE_STRIDE + m * 8;
#pragma unroll
        for (int d = 0; d < 4; ++d) {
            const float re = r00 * wn[0  + d * 2] + r01 * wn[8  + d * 2]
                           + r10 * wn[16 + d * 2] + r11 * wn[24 + d * 2];
            const float im = r00 * wn[1  + d * 2] + r01 * wn[9  + d * 2]
                           + r10 * wn[17 + d * 2] + r11 * wn[25 + d * 2];
            dn[d * 2 + 0] = re;
            dn[d * 2 + 1] = im;
        }
    }
}

// Top contraction: out[b, l] = sum_{i,j} v0[i]*v1[j]*top_w[i,j,l]
__device__ __forceinline__ void top_layer(const float* sw, const float* v2buf,
                                          float* __restrict__ out, int bBase, int lane)
{
    const int m = lane & 15;
    const int l = lane >> 4;             // label 0/1
    const float* vL = v2buf + 0 * NODE_STRIDE + m * 8;
    const float* vR = v2buf + 1 * NODE_STRIDE + m * 8;
    float ore = 0.f, oim = 0.f;
#pragma unroll
    for (int i = 0; i < 4; ++i)
#pragma unroll
        for (int j = 0; j < 4; ++j) {
            const float cre = vL[i * 2] * vR[j * 2]     - vL[i * 2 + 1] * vR[j * 2 + 1];
            const float cim = vL[i * 2] * vR[j * 2 + 1] + vL[i * 2 + 1] * vR[j * 2];
            const float wre = sw[OFF_TOP + ((i * 4 + j) * 2 + l) * 2 + 0];
            const float wim = sw[OFF_TOP + ((i * 4 + j) * 2 + l) * 2 + 1];
            ore += cre * wre - cim * wim;
            oim += cre * wim + cim * wre;
        }
    const size_t o = ((size_t)(bBase + m) * 2 + l) * 2;
    out[o + 0] = ore;
    out[o + 1] = oim;
}

__global__ __launch_bounds__(THREADS) void ttn_fused_kernel(
    const float* __restrict__ x,
    const float* __restrict__ topw,
    const float* __restrict__ m1, const float* __restrict__ m2,
    const float* __restrict__ m3, const float* __restrict__ m4,
    const float* __restrict__ m5, const float* __restrict__ m6,
    const float* __restrict__ leafw,
    float* __restrict__ out)
{
    __shared__ float s_w[W_TOTAL];                    // ~79 KB weights (block-shared)
    __shared__ float s_v[WAVES_PER_BLOCK * VW_SIZE];  // 80 KB value buffers (per-wave)

    const int tid = threadIdx.x;
    // Stage all weights into LDS once (tiny, L2-hot, reused by every tile).
    for (int i = tid; i < 64;   i += THREADS) s_w[OFF_TOP  + i] = topw[i];
    for (int i = tid; i < 256;  i += THREADS) s_w[OFF_M1   + i] = m1[i];
    for (int i = tid; i < 512;  i += THREADS) s_w[OFF_M2   + i] = m2[i];
    for (int i = tid; i < 1024; i += THREADS) s_w[OFF_M3   + i] = m3[i];
    for (int i = tid; i < 2048; i += THREADS) s_w[OFF_M4   + i] = m4[i];
    for (int i = tid; i < 4096; i += THREADS) s_w[OFF_M5   + i] = m5[i];
    for (int i = tid; i < 8192; i += THREADS) s_w[OFF_M6   + i] = m6[i];
    for (int i = tid; i < 4096; i += THREADS) s_w[OFF_LEAF + i] = leafw[i];
    __syncthreads();

    const int wave  = tid >> 5;
    const int lane  = tid & 31;
    const int bBase = (blockIdx.x * WAVES_PER_BLOCK + wave) * SAMP_PER_WAVE;

    float* vw   = s_v + wave * VW_SIZE;
    float* bufA = vw + BUFA;
    float* bufB = vw + BUFB;
    float* v16  = vw + V16;

    // Phase 1: 8 subtrees of 16 leaf nodes each -> 2 nodes of the 16-node layer.
    for (int s = 0; s < 8; ++s) {
        leaf_layer(x, s_w + OFF_LEAF, bufA, bBase, s, lane);       // 16 nodes
        __syncthreads();
        apply_layer(s_w + OFF_M6, bufA, bufB, 8, 8 * s, lane);     // 128 -> 64 layer
        __syncthreads();
        apply_layer(s_w + OFF_M5, bufB, bufA, 4, 4 * s, lane);     // 64 -> 32 layer
        __syncthreads();
        apply_layer(s_w + OFF_M4, bufA, v16 + 2 * s * NODE_STRIDE,
                    2, 2 * s, lane);                               // 32 -> 16 layer
        __syncthreads();
    }

    // Phase 2: 16 -> 8 -> 4 -> 2 -> labels.
    apply_layer(s_w + OFF_M3, v16, bufA, 8, 0, lane);
    __syncthreads();
    apply_layer(s_w + OFF_M2, bufA, bufB, 4, 0, lane);
    __syncthreads();
    apply_layer(s_w + OFF_M1, bufB, bufA, 2, 0, lane);
    __syncthreads();
    top_layer(s_w, bufA, out, bBase, lane);
}

extern "C" void kernel_launch(void* const* d_in, const int* in_sizes, int n_in,
                              void* d_out, int out_size, void* d_ws, size_t ws_size,
                              hipStream_t stream) {
    (void)in_sizes; (void)n_in; (void)out_size; (void)d_ws; (void)ws_size;
    // setup_inputs order: x, top_w, mids[0..5], leaf_w (complex64 -> interleaved float pairs)
    const float* x     = (const float*)d_in[0];
    const float* topw  = (const float*)d_in[1];
    const float* m1    = (const float*)d_in[2];
    const float* m2    = (const float*)d_in[3];
    const float* m3    = (const float*)d_in[4];
    const float* m4    = (const float*)d_in[5];
    const float* m5    = (const float*)d_in[6];
    const float* m6    = (const float*)d_in[7];
    const float* leafw = (const float*)d_in[8];

    dim3 grid(BATCH / (WAVES_PER_BLOCK * SAMP_PER_WAVE));   // 1024 blocks
    dim3 block(THREADS);                                    // 4 waves
    ttn_fused_kernel<<<grid, block, 0, stream>>>(
        x, topw, m1, m2, m3, m4, m5, m6, leafw, (float*)d_out);
}